// MAGNN_Agg_43765716746888
// MI455X (gfx1250) — compile-verified
//
#include <hip/hip_runtime.h>
#include <hip/hip_bf16.h>

typedef __attribute__((ext_vector_type(2))) float v2f;
typedef __attribute__((ext_vector_type(8))) float v8f;

#define DF 128

// ---------------------------------------------------------------------------
// Edge scatter: sum[sidx[e], :] += feat[gidx[e], :] * (w ? w[e] : 1);
//               cnt[sidx[e]]    += 1
// One wave (32 lanes) per edge, each lane owns a float4 slice: coalesced
// b128 gathers, 4 f32 atomics per lane (targets are L2-resident).
// ---------------------------------------------------------------------------
__global__ void edge_scatter_kernel(const float* __restrict__ feat,
                                    const float* __restrict__ w,
                                    const int* __restrict__ gidx,
                                    const int* __restrict__ sidx,
                                    float* __restrict__ sum,
                                    float* __restrict__ cnt, int nE) {
  long long t = (long long)blockIdx.x * blockDim.x + threadIdx.x;
  int e = (int)(t >> 5);
  int q = (int)(t & 31);                       // float4 slot within the row
  if (e >= nE) return;
  int g = gidx[e];
  int s = sidx[e];
  float4 v = ((const float4*)(feat + (size_t)g * DF))[q];
  if (w) {
    float ww = w[e];
    v.x *= ww; v.y *= ww; v.z *= ww; v.w *= ww;
  }
  float* sp = sum + (size_t)s * DF + q * 4;
  atomicAdd(sp + 0, v.x);
  atomicAdd(sp + 1, v.y);
  atomicAdd(sp + 2, v.z);
  atomicAdd(sp + 3, v.w);
  if (q == 0) atomicAdd(cnt + s, 1.0f);
}

// net[t] = (sum[t]/max(cnt,1) + xm[t]) * 0.5      (float4 granularity)
__global__ void combine_mean_kernel(const float4* __restrict__ sum,
                                    const float* __restrict__ cnt,
                                    const float4* __restrict__ xm,
                                    float4* __restrict__ net, int total4) {
  int t = blockIdx.x * blockDim.x + threadIdx.x;
  if (t >= total4) return;
  float c = 1.0f / fmaxf(cnt[t >> 5], 1.0f);
  float4 s = sum[t], x = xm[t], o;
  o.x = (s.x * c + x.x) * 0.5f;
  o.y = (s.y * c + x.y) * 0.5f;
  o.z = (s.z * c + x.z) * 0.5f;
  o.w = (s.w * c + x.w) * 0.5f;
  net[t] = o;
}

// sum[t] /= max(cnt,1)   (in place, float4 granularity)
__global__ void divide_inplace_kernel(float4* __restrict__ sum,
                                      const float* __restrict__ cnt, int total4) {
  int t = blockIdx.x * blockDim.x + threadIdx.x;
  if (t >= total4) return;
  float c = 1.0f / fmaxf(cnt[t >> 5], 1.0f);
  float4 s = sum[t];
  s.x *= c; s.y *= c; s.z *= c; s.w *= c;
  sum[t] = s;
}

// ---------------------------------------------------------------------------
// out[n, p, :] = relu(h[n,:] @ W^T + b)  via V_WMMA_F32_16X16X4_F32 (fp32).
// Block = 256 threads = 8 waves; wave w owns the 16x16 tile at columns w*16.
// A tile: 16x4 fp32, lane m=lane&15, k=(lane>>4)*2+v (ISA layout).
// B tile: 4x16 fp32 = W^T slice, lane n=lane&15, k=(lane>>4)*2+v.
// D/C: VGPR r -> (M = r + (lane>>4)*8, N = lane&15).
// ---------------------------------------------------------------------------
__global__ void __launch_bounds__(256)
gemm_relu_wmma_kernel(const float* __restrict__ h, const float* __restrict__ W,
                      const float* __restrict__ bias, float* __restrict__ outAllm,
                      int p, int n0) {
  int lane = threadIdx.x & 31;
  int wv   = threadIdx.x >> 5;
  int row0 = blockIdx.x * 16;
  int col0 = wv * 16;
  int nl   = lane & 15;
  int khalf = (lane >> 4) * 2;

  int mrow = row0 + nl;
  if (mrow >= n0) mrow = n0 - 1;               // clamp tail loads
  const float* Arow = h + (size_t)mrow * DF + khalf;
  const float* Bcol = W + (size_t)(col0 + nl) * DF + khalf;

  v8f acc = {};
#pragma unroll 8
  for (int k0 = 0; k0 < DF; k0 += 4) {
    v2f a = *(const v2f*)(Arow + k0);          // 8B-aligned (khalf even)
    v2f b = *(const v2f*)(Bcol + k0);
    acc = __builtin_amdgcn_wmma_f32_16x16x4_f32(
        /*neg_a=*/false, a, /*neg_b=*/false, b,
        /*c_mod=*/(short)0, acc, /*reuse_a=*/false, /*reuse_b=*/false);
  }

  float bv = bias[col0 + nl];
  int rbase = (lane >> 4) * 8;
  float* outp = outAllm + (size_t)(row0 + rbase) * (5 * DF) + (size_t)p * DF + col0 + nl;
  if (row0 + 16 <= n0) {                       // hot path: straight-line stores
#pragma unroll
    for (int r = 0; r < 8; ++r)
      outp[(size_t)r * (5 * DF)] = fmaxf(acc[r] + bv, 0.0f);
  } else {
#pragma unroll
    for (int r = 0; r < 8; ++r) {
      int row = row0 + rbase + r;
      if (row < n0)
        outp[(size_t)r * (5 * DF)] = fmaxf(acc[r] + bv, 0.0f);
    }
  }
}

// ---------------------------------------------------------------------------
// Attention: scores[n,p] = dot(allm[n,p,:], att_vec[p,:]); softmax over p;
// out[n,:] = sum_p allm[n,p,:] * att[n,p].  One block (128 thr) per node.
// ---------------------------------------------------------------------------
__global__ void attn_kernel(const float* __restrict__ allm,
                            const float* __restrict__ att_vec,
                            float* __restrict__ out, int n0) {
  int n = blockIdx.x;
  if (n >= n0) return;
  int d = threadIdx.x;                  // 0..127
  int lane = d & 31, wv = d >> 5;
  __shared__ float lred[5][4];

  const float* row = allm + (size_t)n * (5 * DF);
  float v[5];
#pragma unroll
  for (int p = 0; p < 5; ++p) {
    v[p] = row[p * DF + d];
    float s = v[p] * att_vec[p * DF + d];
#pragma unroll
    for (int off = 16; off > 0; off >>= 1) s += __shfl_down(s, off, 32);
    if (lane == 0) lred[p][wv] = s;
  }
  __syncthreads();

  float tot[5], m = -1e30f;
#pragma unroll
  for (int p = 0; p < 5; ++p) {
    tot[p] = lred[p][0] + lred[p][1] + lred[p][2] + lred[p][3];
    m = fmaxf(m, tot[p]);
  }
  float den = 0.0f;
#pragma unroll
  for (int p = 0; p < 5; ++p) { tot[p] = __expf(tot[p] - m); den += tot[p]; }
  float inv = 1.0f / den;
  float o = 0.0f;
#pragma unroll
  for (int p = 0; p < 5; ++p) o += v[p] * tot[p] * inv;
  out[(size_t)n * DF + d] = o;
}

// ---------------------------------------------------------------------------
extern "C" void kernel_launch(void* const* d_in, const int* in_sizes, int n_in,
                              void* d_out, int out_size, void* d_ws, size_t ws_size,
                              hipStream_t stream) {
  const float* x_node = (const float*)d_in[0];
  const float* x1 = (const float*)d_in[1];
  const float* x2 = (const float*)d_in[2];
  const float* x3 = (const float*)d_in[3];
  const float* w1 = (const float*)d_in[4];
  const float* w2 = (const float*)d_in[5];
  const float* w3 = (const float*)d_in[6];
  const float* W1 = (const float*)d_in[7];
  const float* b1 = (const float*)d_in[8];
  const float* W2 = (const float*)d_in[9];
  const float* b2 = (const float*)d_in[10];
  const float* W3 = (const float*)d_in[11];
  const float* b3 = (const float*)d_in[12];
  const float* W121 = (const float*)d_in[13];
  const float* b121 = (const float*)d_in[14];
  const float* W131 = (const float*)d_in[15];
  const float* b131 = (const float*)d_in[16];
  const float* att_vec = (const float*)d_in[17];
  const int* e1_src = (const int*)d_in[18];
  const int* e1_dst = (const int*)d_in[19];
  const int* e2_src = (const int*)d_in[20];
  const int* e2_dst = (const int*)d_in[21];
  const int* e3_src = (const int*)d_in[22];
  const int* e3_dst = (const int*)d_in[23];
  const int* e12_src = (const int*)d_in[24];
  const int* e12_dst = (const int*)d_in[25];
  const int* e13_src = (const int*)d_in[26];
  const int* e13_dst = (const int*)d_in[27];

  const int N0 = in_sizes[0] / DF;
  const int N1 = in_sizes[1] / DF;
  const int N2 = in_sizes[2] / DF;
  const int N3 = in_sizes[3] / DF;
  const int E   = in_sizes[4];
  const int E12 = in_sizes[24];
  int Nm = N1 > N2 ? N1 : N2; if (N3 > Nm) Nm = N3;

  // workspace layout (floats), all offsets multiples of 128 floats (512B)
  float* ws   = (float*)d_ws;
  size_t off  = 0;
  float* allm = ws + off; off += (size_t)N0 * 5 * DF;
  float* net1 = ws + off; off += (size_t)N1 * DF;
  float* sum  = ws + off; off += (size_t)N0 * DF;
  float* cnt  = ws + off; off += ((size_t)N0 + 127) & ~(size_t)127;
  float* nbuf = ws + off; off += (size_t)Nm * DF;
  (void)ws_size;

  auto scatter = [&](const float* feat, const float* w, const int* g,
                     const int* s, int nTgt, int nE) {
    hipMemsetAsync(sum, 0, (size_t)nTgt * DF * sizeof(float), stream);
    hipMemsetAsync(cnt, 0, (size_t)nTgt * sizeof(float), stream);
    long long thr = (long long)nE * 32;        // one wave per edge
    int blocks = (int)((thr + 255) / 256);
    edge_scatter_kernel<<<blocks, 256, 0, stream>>>(feat, w, g, s, sum, cnt, nE);
  };
  auto combine = [&](const float* xm, float* net, int nrows) {
    int total4 = nrows * (DF / 4);
    combine_mean_kernel<<<(total4 + 255) / 256, 256, 0, stream>>>(
        (const float4*)sum, cnt, (const float4*)xm, (float4*)net, total4);
  };
  auto divide = [&](int nrows) {
    int total4 = nrows * (DF / 4);
    divide_inplace_kernel<<<(total4 + 255) / 256, 256, 0, stream>>>(
        (float4*)sum, cnt, total4);
  };
  auto gemm = [&](const float* h, const float* W, const float* b, int p) {
    gemm_relu_wmma_kernel<<<(N0 + 15) / 16, 256, 0, stream>>>(h, W, b, allm, p, N0);
  };

  // ---- short path 1 (also produces net1, shared with long paths) ----
  scatter(x_node, w1, e1_src, e1_dst, N1, E);
  combine(x1, net1, N1);
  scatter(net1, nullptr, e1_dst, e1_src, N0, E);
  divide(N0);
  gemm(sum, W1, b1, 0);

  // ---- short path 2 ----
  scatter(x_node, w2, e2_src, e2_dst, N2, E);
  combine(x2, nbuf, N2);
  scatter(nbuf, nullptr, e2_dst, e2_src, N0, E);
  divide(N0);
  gemm(sum, W2, b2, 1);

  // ---- short path 3 ----
  scatter(x_node, w3, e3_src, e3_dst, N3, E);
  combine(x3, nbuf, N3);
  scatter(nbuf, nullptr, e3_dst, e3_src, N0, E);
  divide(N0);
  gemm(sum, W3, b3, 2);

  // ---- long path s-1-2-1-s ----
  scatter(net1, nullptr, e12_src, e12_dst, N2, E12);
  combine(x2, nbuf, N2);                       // n2
  scatter(nbuf, nullptr, e12_dst, e12_src, N1, E12);
  combine(x1, nbuf, N1);                       // n3 (n2 dead)
  scatter(nbuf, w1, e1_dst, e1_src, N0, E);
  divide(N0);
  gemm(sum, W121, b121, 3);

  // ---- long path s-1-3-1-s ----
  scatter(net1, nullptr, e13_src, e13_dst, N3, E12);
  combine(x3, nbuf, N3);
  scatter(nbuf, nullptr, e13_dst, e13_src, N1, E12);
  combine(x1, nbuf, N1);
  scatter(nbuf, w1, e1_dst, e1_src, N0, E);
  divide(N0);
  gemm(sum, W131, b131, 4);

  // ---- attention combine ----
  attn_kernel<<<N0, DF, 0, stream>>>(allm, att_vec, (float*)d_out, N0);
}